// ResidualGATBlock_34677565948181
// MI455X (gfx1250) — compile-verified
//
#include <hip/hip_runtime.h>
#include <hip/hip_bf16.h>

#define N_NODES 50000
#define N_EDGES 800000
#define IN_CH   64
#define HEADS   4
#define OUT_CH  16
#define DCH     64          // HEADS*OUT_CH == IN_CH
#define NEG_SLOPE 0.2f
#define LN_EPS    1e-5f

typedef __attribute__((ext_vector_type(2))) float v2f;
typedef __attribute__((ext_vector_type(8))) float v8f;

__device__ __forceinline__ float lrelu(float v) {
    return v > 0.0f ? v : NEG_SLOPE * v;
}

// Order-preserving float atomic max via integer punning (safe on all toolchains;
// lowers to global_atomic_max_i32 / global_atomic_min_u32 in L2).
__device__ __forceinline__ void atomicMaxF(float* addr, float v) {
    if (v >= 0.0f) {
        atomicMax((int*)addr, __float_as_int(v));
    } else {
        atomicMin((unsigned int*)addr, (unsigned int)__float_as_int(v));
    }
}

// ---------------------------------------------------------------------------
// 0) workspace init: nmax = -inf, denom = 0, accum = 0
// ---------------------------------------------------------------------------
__global__ __launch_bounds__(256) void gat_init(float* __restrict__ nmax,
                                                float* __restrict__ denom,
                                                float* __restrict__ accum) {
    int t = blockIdx.x * 256 + threadIdx.x;
    if (t < N_NODES * HEADS) {
        nmax[t]  = __int_as_float(0xFF800000);  // -inf
        denom[t] = 0.0f;
    }
    if (t < N_NODES * DCH) accum[t] = 0.0f;
}

// ---------------------------------------------------------------------------
// 1) Dual GEMM with V_WMMA_F32_16X16X4_F32:
//    xl = x@Wl + bl ; xr = x@Wr + br
//    Block = 256 thr = 8 waves. Wave w: (w<4 ? xl : xr), column tile n0=(w&3)*16.
//    Each wave owns one 16x16 output tile; K=64 -> 16 chained WMMAs.
// ---------------------------------------------------------------------------
__global__ __launch_bounds__(256)
void gat_dual_gemm_wmma(const float* __restrict__ x,
                        const float* __restrict__ Wl, const float* __restrict__ bl,
                        const float* __restrict__ Wr, const float* __restrict__ br,
                        float* __restrict__ xl, float* __restrict__ xr) {
    const int lane = threadIdx.x & 31;
    const int wave = threadIdx.x >> 5;
    const int m0   = blockIdx.x * 16;
    const int n0   = (wave & 3) * 16;
    const bool is_r = (wave >= 4);

    const float* __restrict__ W  = is_r ? Wr : Wl;
    const float* __restrict__ bv = is_r ? br : bl;
    float* __restrict__ out      = is_r ? xr : xl;

    const int half = lane >> 4;      // lane half selects K pair {0,1} vs {2,3}
    const int l16  = lane & 15;
    const int row  = m0 + l16;       // A-matrix: M = lane%16
    const int col  = n0 + l16;       // B/C/D:    N = lane%16

    v8f acc = {};
#pragma unroll
    for (int k0 = 0; k0 < IN_CH; k0 += 4) {
        const int ka = k0 + half * 2;
        v2f a, b;
        // A 16x4: VGPR v holds K = ka + v  (doc'd layout)
        a.x = x[row * IN_CH + ka];
        a.y = x[row * IN_CH + ka + 1];
        // B 4x16 (row K striped across lanes per VGPR, symmetric to A)
        b.x = W[ka * DCH + col];
        b.y = W[(ka + 1) * DCH + col];
        acc = __builtin_amdgcn_wmma_f32_16x16x4_f32(
            /*neg_a=*/false, a, /*neg_b=*/false, b,
            /*c_mod=*/(short)0, acc, /*reuse_a=*/false, /*reuse_b=*/false);
    }
    // D layout: VGPR r -> M = r + half*8, N = lane%16
#pragma unroll
    for (int r = 0; r < 8; ++r) {
        const int m = m0 + r + half * 8;
        out[m * DCH + col] = acc[r] + bv[col];
    }
}

// ---------------------------------------------------------------------------
// 2) Per-(edge,head) attention logit + segment max (float atomic max).
//    thread t -> edge e = t/4, head h = t%4; float4-vectorized 16-ch dot.
// ---------------------------------------------------------------------------
__global__ __launch_bounds__(256)
void gat_edge_logits(const int* __restrict__ src, const int* __restrict__ dst,
                     const float* __restrict__ xl, const float* __restrict__ xr,
                     const float* __restrict__ att,
                     float* __restrict__ logit, float* __restrict__ nmax) {
    const int t = blockIdx.x * 256 + threadIdx.x;
    if (t >= N_EDGES * HEADS) return;
    const int e = t >> 2;
    const int h = t & 3;
    const int s = src[e];
    const int d = dst[e];

    const float4* __restrict__ pl = (const float4*)(xl + s * DCH + h * OUT_CH);
    const float4* __restrict__ pr = (const float4*)(xr + d * DCH + h * OUT_CH);
    const float4* __restrict__ pa = (const float4*)(att + h * OUT_CH);

    float acc = 0.0f;
#pragma unroll
    for (int i = 0; i < 4; ++i) {
        const float4 a = pl[i];
        const float4 b = pr[i];
        const float4 w = pa[i];
        acc += w.x * lrelu(a.x + b.x);
        acc += w.y * lrelu(a.y + b.y);
        acc += w.z * lrelu(a.z + b.z);
        acc += w.w * lrelu(a.w + b.w);
    }
    logit[t] = acc;
    atomicMaxF(&nmax[d * HEADS + h], acc);
}

// ---------------------------------------------------------------------------
// 3) a_exp = exp(logit - max[dst]) (in place) + segment-sum denominator.
// ---------------------------------------------------------------------------
__global__ __launch_bounds__(256)
void gat_edge_exp(const int* __restrict__ dst,
                  float* __restrict__ logit,           // in: logit, out: a_exp
                  const float* __restrict__ nmax,
                  float* __restrict__ denom) {
    const int t = blockIdx.x * 256 + threadIdx.x;
    if (t >= N_EDGES * HEADS) return;
    const int e = t >> 2;
    const int h = t & 3;
    const int d = dst[e];
    const float a = __expf(logit[t] - nmax[d * HEADS + h]);
    logit[t] = a;
    atomicAdd(&denom[d * HEADS + h], a);
}

// ---------------------------------------------------------------------------
// 4) Weighted scatter-aggregate: accum[dst] += alpha * xl[src].
//    16 threads per edge, one float4 (4 channels) each.
// ---------------------------------------------------------------------------
__global__ __launch_bounds__(256)
void gat_edge_scatter(const int* __restrict__ src, const int* __restrict__ dst,
                      const float* __restrict__ xl,
                      const float* __restrict__ aexp,
                      const float* __restrict__ denom,
                      float* __restrict__ accum) {
    const int t = blockIdx.x * 256 + threadIdx.x;
    if (t >= N_EDGES * 16) return;
    const int e = t >> 4;
    const int j = t & 15;
    const int h = j >> 2;
    const int q = (j & 3) * 4;
    const int s = src[e];
    const int d = dst[e];
    const float alpha = aexp[e * HEADS + h] / denom[d * HEADS + h];
    const float4 v = *(const float4*)(xl + s * DCH + h * OUT_CH + q);
    float* __restrict__ o = accum + d * DCH + h * OUT_CH + q;
    atomicAdd(o + 0, alpha * v.x);
    atomicAdd(o + 1, alpha * v.y);
    atomicAdd(o + 2, alpha * v.z);
    atomicAdd(o + 3, alpha * v.w);
}

// ---------------------------------------------------------------------------
// 5) Residual + LayerNorm + ELU. One wave32 per node, 2 channels per lane,
//    butterfly __shfl_xor reduction for mean/var.
// ---------------------------------------------------------------------------
__global__ __launch_bounds__(256)
void gat_finalize(const float* __restrict__ accum, const float* __restrict__ x,
                  const float* __restrict__ bias, const float* __restrict__ gamma,
                  const float* __restrict__ beta, float* __restrict__ out) {
    const int gwave = (blockIdx.x * 256 + threadIdx.x) >> 5;
    const int lane  = threadIdx.x & 31;
    if (gwave >= N_NODES) return;
    const int c0 = lane;
    const int c1 = lane + 32;
    const int base = gwave * DCH;

    float v0 = accum[base + c0] + bias[c0] + x[base + c0];
    float v1 = accum[base + c1] + bias[c1] + x[base + c1];

    float s  = v0 + v1;
    float s2 = v0 * v0 + v1 * v1;
#pragma unroll
    for (int off = 16; off > 0; off >>= 1) {
        s  += __shfl_xor(s, off, 32);
        s2 += __shfl_xor(s2, off, 32);
    }
    const float mu  = s * (1.0f / DCH);
    const float var = s2 * (1.0f / DCH) - mu * mu;
    const float inv = rsqrtf(var + LN_EPS);

    float o0 = (v0 - mu) * inv * gamma[c0] + beta[c0];
    float o1 = (v1 - mu) * inv * gamma[c1] + beta[c1];
    out[base + c0] = o0 > 0.0f ? o0 : expm1f(o0);
    out[base + c1] = o1 > 0.0f ? o1 : expm1f(o1);
}

// ---------------------------------------------------------------------------
// launch
// ---------------------------------------------------------------------------
extern "C" void kernel_launch(void* const* d_in, const int* in_sizes, int n_in,
                              void* d_out, int out_size, void* d_ws, size_t ws_size,
                              hipStream_t stream) {
    const float* x   = (const float*)d_in[0];
    const int*   ei  = (const int*)d_in[1];        // [2, N_EDGES]
    const float* Wl  = (const float*)d_in[2];
    const float* bl  = (const float*)d_in[3];
    const float* Wr  = (const float*)d_in[4];
    const float* br  = (const float*)d_in[5];
    const float* att = (const float*)d_in[6];
    const float* bias  = (const float*)d_in[7];
    const float* gamma = (const float*)d_in[8];
    const float* beta  = (const float*)d_in[9];
    float* out = (float*)d_out;

    const int* src = ei;
    const int* dst = ei + N_EDGES;

    // workspace partition (floats); all chunk sizes are multiples of 16B.
    float* ws    = (float*)d_ws;
    float* xl    = ws;                         // N*64
    float* xr    = xl + N_NODES * DCH;         // N*64
    float* accum = xr + N_NODES * DCH;         // N*64
    float* logit = accum + N_NODES * DCH;      // E*4 (reused for a_exp)
    float* nmax  = logit + N_EDGES * HEADS;    // N*4
    float* denom = nmax + N_NODES * HEADS;     // N*4

    // 0) init scatter targets
    gat_init<<<(N_NODES * DCH + 255) / 256, 256, 0, stream>>>(nmax, denom, accum);

    // 1) xl / xr via WMMA (50000 rows = 3125 * 16)
    gat_dual_gemm_wmma<<<N_NODES / 16, 256, 0, stream>>>(x, Wl, bl, Wr, br, xl, xr);

    // 2) logits + segment max
    gat_edge_logits<<<(N_EDGES * HEADS + 255) / 256, 256, 0, stream>>>(
        src, dst, xl, xr, att, logit, nmax);

    // 3) exp + denominator
    gat_edge_exp<<<(N_EDGES * HEADS + 255) / 256, 256, 0, stream>>>(
        dst, logit, nmax, denom);

    // 4) weighted scatter aggregate
    gat_edge_scatter<<<(N_EDGES * 16 + 255) / 256, 256, 0, stream>>>(
        src, dst, xl, logit, denom, accum);

    // 5) residual + LN + ELU
    gat_finalize<<<(N_NODES + 7) / 8, 256, 0, stream>>>(
        accum, x, bias, gamma, beta, out);
}